// First_n_Last_graphER_66855460929562
// MI455X (gfx1250) — compile-verified
//
#include <hip/hip_runtime.h>

typedef float v2f __attribute__((ext_vector_type(2)));
typedef float v8f __attribute__((ext_vector_type(8)));

#define S_LEN    2048
#define HDIM     256
#define TWO_H    512
#define FFN      768
#define K_SPANS  16384      // spans per batch
#define M_TILE   32         // spans per block (2 WMMA M-tiles)
#define A_STRIDE 516        // 512 + 4 pad  (stride % 64 == 4 -> conflict-free b64 reads)
#define H_STRIDE 772        // 768 + 4 pad
#define NT1      6          // N-tiles per wave in GEMM1 (48 tiles / 8 waves)
#define NT2      2          // N-tiles per wave in GEMM2 (16 tiles / 8 waves)

#define W1_PAIRS ((TWO_H / 2) * FFN)     // 196608
#define W2_PAIRS ((FFN / 2) * HDIM)      //  98304

// Pack weights K-pair-interleaved: Wp[k2*N + col] = {W[2k2][col], W[2k2+1][col]}
// -> B-fragment for WMMA becomes a single global_load_b64.
__global__ __launch_bounds__(256) void pack_weights_kernel(
    const float* __restrict__ W1, const float* __restrict__ W2,
    float2* __restrict__ Wp1, float2* __restrict__ Wp2)
{
    int idx = blockIdx.x * 256 + threadIdx.x;
    if (idx < W1_PAIRS) {
        int k2 = idx / FFN, col = idx - k2 * FFN;
        float2 p;
        p.x = W1[(size_t)(2 * k2)     * FFN + col];
        p.y = W1[(size_t)(2 * k2 + 1) * FFN + col];
        Wp1[idx] = p;
    }
    if (idx < W2_PAIRS) {
        int k2 = idx >> 8, col = idx & (HDIM - 1);
        float2 p;
        p.x = W2[(size_t)(2 * k2)     * HDIM + col];
        p.y = W2[(size_t)(2 * k2 + 1) * HDIM + col];
        Wp2[idx] = p;
    }
}

template<bool PACKED>
__global__ __launch_bounds__(256) void span_ffn_kernel(
    const float* __restrict__ token_reps,
    const long long* __restrict__ span_ids,
    const unsigned char* __restrict__ span_masks,
    const float* __restrict__ W1,
    const float* __restrict__ b1,
    const float* __restrict__ W2,
    const float* __restrict__ b2,
    const float2* __restrict__ Wp1,
    const float2* __restrict__ Wp2,
    float* __restrict__ out)
{
    extern __shared__ float smem[];
    float* Arep = smem;                          // [M_TILE][A_STRIDE]
    float* Hmid = smem + M_TILE * A_STRIDE;      // [M_TILE][H_STRIDE]

    const int tid  = threadIdx.x;
    const int lane = tid & 31;
    const int wave = tid >> 5;
    const int l15  = lane & 15;
    const int half = lane >> 4;                  // 0: lanes 0-15, 1: lanes 16-31
    const int spanBase = blockIdx.x * M_TILE;
    const int batch = spanBase / K_SPANS;        // 32 | K_SPANS -> uniform per block
    const float* __restrict__ tr = token_reps + (size_t)batch * S_LEN * HDIM;

    // ---------------- Phase 0: gather + relu -> Arep (float4 vectorized) ----
    for (int i = tid; i < M_TILE * (TWO_H / 4); i += 256) {
        int row = i >> 7;            // 0..31
        int c4  = (i & 127) << 2;    // 0,4,...,508
        int span = spanBase + row;
        long long m  = (long long)span_masks[span];
        long long s0 = span_ids[2 * span] * m;
        long long s1 = span_ids[2 * span + 1] * m - 1;
        long long id = (c4 < HDIM) ? s0 : s1;
        if (id < 0) id = 0;
        if (id > S_LEN - 1) id = S_LEN - 1;
        const float4* src = (const float4*)(tr + (size_t)id * HDIM + (c4 & (HDIM - 1)));
        float4 v = *src;
        v.x = v.x > 0.f ? v.x : 0.f;
        v.y = v.y > 0.f ? v.y : 0.f;
        v.z = v.z > 0.f ? v.z : 0.f;
        v.w = v.w > 0.f ? v.w : 0.f;
        *(float4*)(Arep + row * A_STRIDE + c4) = v;
    }
    __syncthreads();

    // ---------------- Phase 1: GEMM1 (32x512 @ 512x768) + bias + relu -> Hmid
    {
        v8f acc[2][NT1];
        #pragma unroll
        for (int mt = 0; mt < 2; ++mt)
            #pragma unroll
            for (int j = 0; j < NT1; ++j)
                acc[mt][j] = (v8f){0.f,0.f,0.f,0.f,0.f,0.f,0.f,0.f};

        #pragma unroll 2
        for (int k = 0; k < TWO_H; k += 4) {
            int kk = k + (half << 1);
            v2f a0, a1;
            a0.x = Arep[(l15)      * A_STRIDE + kk];
            a0.y = Arep[(l15)      * A_STRIDE + kk + 1];
            a1.x = Arep[(16 + l15) * A_STRIDE + kk];
            a1.y = Arep[(16 + l15) * A_STRIDE + kk + 1];
            #pragma unroll
            for (int j = 0; j < NT1; ++j) {
                int col = (wave * NT1 + j) * 16 + l15;
                v2f bq;
                if (PACKED) {
                    int k2 = (k >> 1) + half;
                    float2 p = Wp1[(size_t)k2 * FFN + col];
                    bq.x = p.x; bq.y = p.y;
                } else {
                    bq.x = W1[(size_t)kk * FFN + col];
                    bq.y = W1[(size_t)(kk + 1) * FFN + col];
                }
                acc[0][j] = __builtin_amdgcn_wmma_f32_16x16x4_f32(
                    false, a0, false, bq, (short)0, acc[0][j], false, false);
                acc[1][j] = __builtin_amdgcn_wmma_f32_16x16x4_f32(
                    false, a1, false, bq, (short)0, acc[1][j], false, false);
            }
        }

        // bias + relu, write to Hmid
        #pragma unroll
        for (int j = 0; j < NT1; ++j) {
            int col  = (wave * NT1 + j) * 16 + l15;
            float bs = b1[col];
            #pragma unroll
            for (int i = 0; i < 8; ++i) {
                int rowm = (half ? 8 : 0) + i;
                float v0 = acc[0][j][i] + bs;
                float v1 = acc[1][j][i] + bs;
                Hmid[rowm        * H_STRIDE + col] = v0 > 0.f ? v0 : 0.f;
                Hmid[(16 + rowm) * H_STRIDE + col] = v1 > 0.f ? v1 : 0.f;
            }
        }
    }
    __syncthreads();

    // ---------------- Phase 2: GEMM2 (32x768 @ 768x256) + bias -> out
    {
        v8f acc[2][NT2];
        #pragma unroll
        for (int mt = 0; mt < 2; ++mt)
            #pragma unroll
            for (int j = 0; j < NT2; ++j)
                acc[mt][j] = (v8f){0.f,0.f,0.f,0.f,0.f,0.f,0.f,0.f};

        #pragma unroll 2
        for (int k = 0; k < FFN; k += 4) {
            int kk = k + (half << 1);
            v2f a0, a1;
            a0.x = Hmid[(l15)      * H_STRIDE + kk];
            a0.y = Hmid[(l15)      * H_STRIDE + kk + 1];
            a1.x = Hmid[(16 + l15) * H_STRIDE + kk];
            a1.y = Hmid[(16 + l15) * H_STRIDE + kk + 1];
            #pragma unroll
            for (int j = 0; j < NT2; ++j) {
                int col = (wave * NT2 + j) * 16 + l15;
                v2f bq;
                if (PACKED) {
                    int k2 = (k >> 1) + half;
                    float2 p = Wp2[(size_t)k2 * HDIM + col];
                    bq.x = p.x; bq.y = p.y;
                } else {
                    bq.x = W2[(size_t)kk * HDIM + col];
                    bq.y = W2[(size_t)(kk + 1) * HDIM + col];
                }
                acc[0][j] = __builtin_amdgcn_wmma_f32_16x16x4_f32(
                    false, a0, false, bq, (short)0, acc[0][j], false, false);
                acc[1][j] = __builtin_amdgcn_wmma_f32_16x16x4_f32(
                    false, a1, false, bq, (short)0, acc[1][j], false, false);
            }
        }

        // bias + store
        #pragma unroll
        for (int j = 0; j < NT2; ++j) {
            int col  = (wave * NT2 + j) * 16 + l15;
            float bs = b2[col];
            #pragma unroll
            for (int i = 0; i < 8; ++i) {
                int rowm = (half ? 8 : 0) + i;
                size_t r0 = (size_t)(spanBase + rowm) * HDIM + col;
                size_t r1 = (size_t)(spanBase + 16 + rowm) * HDIM + col;
                out[r0] = acc[0][j][i] + bs;
                out[r1] = acc[1][j][i] + bs;
            }
        }
    }
}

extern "C" void kernel_launch(void* const* d_in, const int* in_sizes, int n_in,
                              void* d_out, int out_size, void* d_ws, size_t ws_size,
                              hipStream_t stream) {
    const float*         token_reps = (const float*)d_in[0];
    const long long*     span_ids   = (const long long*)d_in[1];
    const unsigned char* span_masks = (const unsigned char*)d_in[2];
    const float*         W1         = (const float*)d_in[3];
    const float*         b1         = (const float*)d_in[4];
    const float*         W2         = (const float*)d_in[5];
    const float*         b2         = (const float*)d_in[6];
    float*               out        = (float*)d_out;

    const int totalSpans = 4 * K_SPANS;                 // B * K = 65536
    const int nBlocks = totalSpans / M_TILE;            // 2048
    const size_t ldsBytes = (size_t)(M_TILE * A_STRIDE + M_TILE * H_STRIDE) * sizeof(float);

    const size_t wsNeeded = (size_t)(W1_PAIRS + W2_PAIRS) * sizeof(float2); // ~2.25 MB

    if (d_ws != nullptr && ws_size >= wsNeeded) {
        float2* Wp1 = (float2*)d_ws;
        float2* Wp2 = Wp1 + W1_PAIRS;
        const int packBlocks = (W1_PAIRS + 255) / 256;  // covers W2_PAIRS too
        pack_weights_kernel<<<packBlocks, 256, 0, stream>>>(W1, W2, Wp1, Wp2);
        span_ffn_kernel<true><<<nBlocks, 256, ldsBytes, stream>>>(
            token_reps, span_ids, span_masks, W1, b1, W2, b2, Wp1, Wp2, out);
    } else {
        span_ffn_kernel<false><<<nBlocks, 256, ldsBytes, stream>>>(
            token_reps, span_ids, span_masks, W1, b1, W2, b2, nullptr, nullptr, out);
    }
}